// AttnDecoderRNN_21449066676577
// MI455X (gfx1250) — compile-verified
//
#include <hip/hip_runtime.h>
#include <math.h>

#define H 1024
#define V 50000
#define E 602
#define S 400
#define VE (V + E)

typedef float v2f __attribute__((ext_vector_type(2)));
typedef float v8f __attribute__((ext_vector_type(8)));
typedef int v4i __attribute__((ext_vector_type(4)));

typedef __attribute__((address_space(1))) v4i gv4i_t;  // global b128 chunk
typedef __attribute__((address_space(3))) v4i lv4i_t;  // LDS b128 chunk

#if __has_builtin(__builtin_amdgcn_global_load_async_to_lds_b128)
#define HAVE_ASYNC_LDS 1
#endif

// Stage nfloat (multiple of 4) floats from global into LDS, whole block.
// Uses the CDNA5 async memory->LDS path (ASYNCcnt) when available.
// Caller must __syncthreads() afterwards.
__device__ __forceinline__ void stage_lds(float* dst, const float* src,
                                          int nfloat) {
  const int tid = threadIdx.x;
  const int nthr = blockDim.x;
#ifdef HAVE_ASYNC_LDS
  unsigned loff = (unsigned)(unsigned long long)(void*)dst;
  unsigned long long gaddr = (unsigned long long)(const void*)src;
  for (int i = tid * 4; i < nfloat; i += nthr * 4) {
    gv4i_t* gp = (gv4i_t*)(gaddr + (unsigned long long)i * 4ull);
    lv4i_t* lp = (lv4i_t*)(unsigned long long)(loff + (unsigned)i * 4u);
    __builtin_amdgcn_global_load_async_to_lds_b128(gp, lp, 0, 0);
  }
#if __has_builtin(__builtin_amdgcn_s_wait_asynccnt)
  __builtin_amdgcn_s_wait_asynccnt(0);
#else
  asm volatile("s_wait_asynccnt 0x0" ::: "memory");
#endif
#else
  for (int i = tid * 4; i < nfloat; i += nthr * 4)
    *(float4*)(dst + i) = *(const float4*)(src + i);
#endif
}

__device__ __forceinline__ float wave_sum(float v) {
#pragma unroll
  for (int off = 16; off > 0; off >>= 1) v += __shfl_down(v, off, 32);
  return v;
}

__device__ __forceinline__ float sigmoidf(float x) {
  return 1.0f / (1.0f + __expf(-x));
}

// ---------------- GRU gate GEMVs: gi = W_ih @ [emb;trigger], gh = W_hh @ h --
// One wave per output row. Rows 0..3071 -> gi, 3072..6143 -> gh.
__global__ void k_gru_gates(const float* __restrict__ W_ih,
                            const float* __restrict__ W_hh,
                            const float* __restrict__ b_ih,
                            const float* __restrict__ b_hh,
                            const float* __restrict__ emb,
                            const int* __restrict__ idx,
                            const float* __restrict__ trigger,
                            const float* __restrict__ hidden,
                            float* __restrict__ gi, float* __restrict__ gh) {
  const int lane = threadIdx.x & 31;
  const int w = (blockIdx.x * blockDim.x + threadIdx.x) >> 5;
  const float* __restrict__ embrow = emb + (size_t)idx[0] * H;
  float sum = 0.0f;
  if (w < 3 * H) {
    const float* __restrict__ wr = W_ih + (size_t)w * (2 * H);
    for (int j = lane * 4; j < H; j += 128) {
      float4 a = *(const float4*)(wr + j);
      float4 b = *(const float4*)(embrow + j);
      sum = fmaf(a.x, b.x, fmaf(a.y, b.y, fmaf(a.z, b.z, fmaf(a.w, b.w, sum))));
    }
    for (int j = lane * 4; j < H; j += 128) {
      float4 a = *(const float4*)(wr + H + j);
      float4 b = *(const float4*)(trigger + j);
      sum = fmaf(a.x, b.x, fmaf(a.y, b.y, fmaf(a.z, b.z, fmaf(a.w, b.w, sum))));
    }
    sum = wave_sum(sum);
    if (lane == 0) gi[w] = sum + b_ih[w];
  } else if (w < 6 * H) {
    const int r = w - 3 * H;
    const float* __restrict__ wr = W_hh + (size_t)r * H;
    for (int j = lane * 4; j < H; j += 128) {
      float4 a = *(const float4*)(wr + j);
      float4 b = *(const float4*)(hidden + j);
      sum = fmaf(a.x, b.x, fmaf(a.y, b.y, fmaf(a.z, b.z, fmaf(a.w, b.w, sum))));
    }
    sum = wave_sum(sum);
    if (lane == 0) gh[r] = sum + b_hh[r];
  }
}

// ---------------- GRU gate combine (elementwise, 1 block of H) -------------
__global__ void k_gru_combine(const float* __restrict__ gi,
                              const float* __restrict__ gh,
                              const float* __restrict__ hidden,
                              float* __restrict__ h_new,
                              float* __restrict__ h_out) {
  const int i = threadIdx.x;
  float r = sigmoidf(gi[i] + gh[i]);
  float z = sigmoidf(gi[H + i] + gh[H + i]);
  float n = tanhf(gi[2 * H + i] + r * gh[2 * H + i]);
  float hv = (1.0f - z) * n + z * hidden[i];
  h_new[i] = hv;
  h_out[i] = hv;  // second tuple output
}

// ---------------- Generic wave-per-row fp32 GEMV (cols % 128 == 0) ---------
__global__ void k_gemv(const float* __restrict__ Wm, const float* __restrict__ x,
                       const float* __restrict__ bias, float* __restrict__ y,
                       int rows, int cols, int relu) {
  const int lane = threadIdx.x & 31;
  const int row = (blockIdx.x * blockDim.x + threadIdx.x) >> 5;
  if (row >= rows) return;
  const float* __restrict__ wr = Wm + (size_t)row * cols;
  float sum = 0.0f;
  for (int j = lane * 4; j < cols; j += 128) {
    float4 a = *(const float4*)(wr + j);
    float4 b = *(const float4*)(x + j);
    sum = fmaf(a.x, b.x, fmaf(a.y, b.y, fmaf(a.z, b.z, fmaf(a.w, b.w, sum))));
  }
  sum = wave_sum(sum);
  if (lane == 0) {
    if (bias) sum += bias[row];
    if (relu) sum = fmaxf(sum, 0.0f);
    y[row] = sum;
  }
}

// ---------------- softmax over S=400 attention scores ----------------------
__global__ void k_softmax_s(const float* __restrict__ scores,
                            float* __restrict__ w, float* __restrict__ w_out) {
  __shared__ float red[512];
  __shared__ float bM, bZ;
  const int t = threadIdx.x;
  float m = -1e30f;
  for (int i = t; i < S; i += 512) m = fmaxf(m, scores[i]);
  red[t] = m;
  __syncthreads();
  for (int s2 = 256; s2 > 0; s2 >>= 1) {
    if (t < s2) red[t] = fmaxf(red[t], red[t + s2]);
    __syncthreads();
  }
  if (t == 0) bM = red[0];
  __syncthreads();
  float s = 0.0f;
  for (int i = t; i < S; i += 512) s += __expf(scores[i] - bM);
  red[t] = s;
  __syncthreads();
  for (int s2 = 256; s2 > 0; s2 >>= 1) {
    if (t < s2) red[t] += red[t + s2];
    __syncthreads();
  }
  if (t == 0) bZ = red[0];
  __syncthreads();
  const float inv = 1.0f / bZ;
  for (int i = t; i < S; i += 512) {
    float v = __expf(scores[i] - bM) * inv;
    w[i] = v;
    w_out[i] = v;  // third tuple output
  }
}

// ---------------- attn_applied[h] = sum_s w[s]*enc[s,h] (1 block of H) -----
__global__ void k_attn_applied(const float* __restrict__ w,
                               const float* __restrict__ enc,
                               float* __restrict__ aa) {
  __shared__ float sw[S];
  const int t = threadIdx.x;
  stage_lds(sw, w, S);  // async memory->LDS path
  __syncthreads();
  float acc = 0.0f;
  for (int s = 0; s < S; ++s) acc = fmaf(sw[s], enc[(size_t)s * H + t], acc);
  aa[t] = acc;
}

// ---------------- p_gen gate + pointer distribution log(atten_p) -----------
__global__ void k_pgen_pointer(const float* __restrict__ aa,
                               const float* __restrict__ h_new,
                               const float* __restrict__ emb,
                               const int* __restrict__ idx,
                               const float* __restrict__ wh_W,
                               const float* __restrict__ ws_W,
                               const float* __restrict__ wx_W,
                               const float* __restrict__ wx_b,
                               const float* __restrict__ attnw,
                               const float* __restrict__ pg_mat,
                               float* __restrict__ pgen_out,
                               float* __restrict__ out) {
  __shared__ float red[H];
  __shared__ float sw[S];
  __shared__ float spg;
  const int t = threadIdx.x;
  const float* __restrict__ embrow = emb + (size_t)idx[0] * H;
  stage_lds(sw, attnw, S);  // async memory->LDS path
  red[t] = wh_W[t] * aa[t] + ws_W[t] * h_new[t] + wx_W[t] * embrow[t];
  __syncthreads();
  for (int s2 = H / 2; s2 > 0; s2 >>= 1) {
    if (t < s2) red[t] += red[t + s2];
    __syncthreads();
  }
  if (t == 0) {
    spg = sigmoidf(red[0] + wx_b[0]);
    pgen_out[0] = spg;
  }
  __syncthreads();
  const float one_m = 1.0f - spg;
  if (t < E) {
    float acc = 0.0f;
    for (int s = 0; s < S; ++s) acc = fmaf(sw[s], pg_mat[(size_t)s * E + t], acc);
    out[V + t] = __logf(acc * one_m);
  }
}

// ---------------- DOMINANT KERNEL: logits = out_W @ ff + out_b via WMMA ----
// One wave per 16-row tile. A = 16x4 fp32 tile of out_W:
//   lanes 0-15  hold rows M=0..15, K = {k, k+1}   (w4.x, w4.y)
//   lanes 16-31 hold rows M=0..15, K = {k+2, k+3} (w4.z, w4.w)
// B replicates ff[k..k+3] across all 16 N columns, so every column of D
// equals the output tile; lane 0 holds y[r0..r0+7], lane 16 y[r0+8..r0+15].
__global__ void k_out_gemv_wmma(const float* __restrict__ Wm,
                                const float* __restrict__ ff,
                                const float* __restrict__ bias,
                                float* __restrict__ y, int rows, int cols) {
  __shared__ float sff[H];
  stage_lds(sff, ff, cols);  // async memory->LDS path
  __syncthreads();

  const int lane = threadIdx.x & 31;
  const int wave = threadIdx.x >> 5;
  const int tile = blockIdx.x * (blockDim.x >> 5) + wave;  // 16-row tile id
  const int r0 = tile * 16;
  if (r0 >= rows) return;  // wave-uniform: EXEC stays all-ones for WMMA

  const bool hi = lane >= 16;
  const float* __restrict__ wp = Wm + (size_t)(r0 + (lane & 15)) * cols;

  v8f c = {};
  for (int k0 = 0; k0 < cols; k0 += 32) {
    __builtin_prefetch(wp + k0 + 256, 0, 1);  // global_prefetch_b8, ~1KB ahead
#pragma unroll
    for (int k = k0; k < k0 + 32; k += 4) {
      float4 w4 = *(const float4*)(wp + k);
      float4 f4 = *(const float4*)(sff + k);  // one ds_load_b128
      v2f a, b;
      a.x = hi ? w4.z : w4.x;
      a.y = hi ? w4.w : w4.y;
      b.x = hi ? f4.z : f4.x;
      b.y = hi ? f4.w : f4.y;
      c = __builtin_amdgcn_wmma_f32_16x16x4_f32(false, a, false, b, (short)0, c,
                                                false, false);
    }
  }
  if (lane == 0) {
#pragma unroll
    for (int j = 0; j < 8; ++j) y[r0 + j] = c[j] + bias[r0 + j];
  } else if (lane == 16) {
#pragma unroll
    for (int j = 0; j < 8; ++j) y[r0 + 8 + j] = c[j] + bias[r0 + 8 + j];
  }
}

// ---------------- vocab softmax: per-block online (max, sumexp) ------------
__global__ void k_vsm_part(const float* __restrict__ logits,
                           float* __restrict__ pairs) {
  __shared__ float rm[256], rs[256];
  const int t = threadIdx.x;
  float m = -1e30f, s = 0.0f;
  for (int i = blockIdx.x * 256 + t; i < V; i += gridDim.x * 256) {
    float x = logits[i];
    float nm = fmaxf(m, x);
    s = s * __expf(m - nm) + __expf(x - nm);
    m = nm;
  }
  rm[t] = m;
  rs[t] = s;
  __syncthreads();
  for (int s2 = 128; s2 > 0; s2 >>= 1) {
    if (t < s2) {
      float m2 = rm[t + s2], v2 = rs[t + s2];
      float M = fmaxf(rm[t], m2);
      rs[t] = rs[t] * __expf(rm[t] - M) + v2 * __expf(m2 - M);
      rm[t] = M;
    }
    __syncthreads();
  }
  if (t == 0) {
    pairs[blockIdx.x * 2 + 0] = rm[0];
    pairs[blockIdx.x * 2 + 1] = rs[0];
  }
}

__global__ void k_vsm_final(const float* __restrict__ pairs,
                            float* __restrict__ mz) {
  __shared__ float rm[64], rs[64];
  const int t = threadIdx.x;
  rm[t] = pairs[t * 2 + 0];
  rs[t] = pairs[t * 2 + 1];
  __syncthreads();
  for (int s2 = 32; s2 > 0; s2 >>= 1) {
    if (t < s2) {
      float m2 = rm[t + s2], v2 = rs[t + s2];
      float M = fmaxf(rm[t], m2);
      rs[t] = rs[t] * __expf(rm[t] - M) + v2 * __expf(m2 - M);
      rm[t] = M;
    }
    __syncthreads();
  }
  if (t == 0) {
    mz[0] = rm[0];
    mz[1] = __logf(rs[0]);
  }
}

// out[i] = log(softmax(l)_i * p_gen) = l_i - M - logZ + log(p_gen)
__global__ void k_write_out(const float* __restrict__ logits,
                            const float* __restrict__ mz,
                            const float* __restrict__ pgen,
                            float* __restrict__ out) {
  const int i = blockIdx.x * blockDim.x + threadIdx.x;
  if (i < V) out[i] = logits[i] - mz[0] - mz[1] + __logf(pgen[0]);
}

extern "C" void kernel_launch(void* const* d_in, const int* in_sizes, int n_in,
                              void* d_out, int out_size, void* d_ws,
                              size_t ws_size, hipStream_t stream) {
  const int* idx = (const int*)d_in[0];
  const float* hidden = (const float*)d_in[1];
  const float* enc = (const float*)d_in[2];
  const float* trigger = (const float*)d_in[3];
  const float* pg_mat = (const float*)d_in[4];
  const float* emb = (const float*)d_in[5];
  const float* attn_W = (const float*)d_in[6];
  const float* comb_W = (const float*)d_in[7];
  const float* comb_b = (const float*)d_in[8];
  const float* W_ih = (const float*)d_in[9];
  const float* W_hh = (const float*)d_in[10];
  const float* b_ih = (const float*)d_in[11];
  const float* b_hh = (const float*)d_in[12];
  const float* out_W = (const float*)d_in[13];
  const float* out_b = (const float*)d_in[14];
  const float* wh_W = (const float*)d_in[15];
  const float* ws_W = (const float*)d_in[16];
  const float* wx_W = (const float*)d_in[17];
  const float* wx_b = (const float*)d_in[18];
  float* out = (float*)d_out;
  float* ws = (float*)d_ws;

  // workspace layout (floats)
  float* gi = ws + 0;                 // 3072
  float* gh = ws + 3 * H;             // 3072
  float* h_new = ws + 6 * H;          // 1024
  float* aa = ws + 7 * H;             // 1024 (contiguous w/ h_new => combined)
  float* q = ws + 8 * H;              // 1024
  float* scores = ws + 9 * H;         // 400
  float* attnw = ws + 9 * H + 512;    // 400
  float* ff = ws + 10 * H;            // 1024
  float* pgen = ws + 11 * H;          // 1
  float* pairs = ws + 11 * H + 32;    // 128
  float* mz = ws + 11 * H + 192;      // 2
  float* logits = ws + 11 * H + 256;  // 50000
  float* combined = h_new;            // [h_new ; attn_applied], 2048

  // 1) GRU gate GEMVs (gi: 3072x2048, gh: 3072x1024) — 6144 waves
  k_gru_gates<<<768, 256, 0, stream>>>(W_ih, W_hh, b_ih, b_hh, emb, idx,
                                       trigger, hidden, gi, gh);
  // 2) gates -> h_new (also 2nd output)
  k_gru_combine<<<1, H, 0, stream>>>(gi, gh, hidden, h_new, out + VE);
  // 3) q = attn_W @ h_new
  k_gemv<<<128, 256, 0, stream>>>(attn_W, h_new, nullptr, q, H, H, 0);
  // 4) scores = enc @ q  (400 rows)
  k_gemv<<<50, 256, 0, stream>>>(enc, q, nullptr, scores, S, H, 0);
  // 5) softmax over scores (also 3rd output)
  k_softmax_s<<<1, 512, 0, stream>>>(scores, attnw, out + VE + H);
  // 6) attn_applied = attnw @ enc
  k_attn_applied<<<1, H, 0, stream>>>(attnw, enc, aa);
  // 7) p_gen + pointer distribution -> out[V .. V+E)
  k_pgen_pointer<<<1, H, 0, stream>>>(aa, h_new, emb, idx, wh_W, ws_W, wx_W,
                                      wx_b, attnw, pg_mat, pgen, out);
  // 8) ff = relu(comb_W @ [h_new; attn_applied] + comb_b)
  k_gemv<<<128, 256, 0, stream>>>(comb_W, combined, comb_b, ff, H, 2 * H, 1);
  // 9) logits = out_W @ ff + out_b   (dominant; WMMA f32 16x16x4 path)
  //    50000/16 = 3125 tiles, 4 waves/block -> 782 blocks
  k_out_gemv_wmma<<<782, 128, 0, stream>>>(out_W, ff, out_b, logits, V, H);
  // 10-11) two-level online softmax stats over V
  k_vsm_part<<<64, 256, 0, stream>>>(logits, pairs);
  k_vsm_final<<<1, 64, 0, stream>>>(pairs, mz);
  // 12) out[0..V) = logits - M - logZ + log(p_gen)
  k_write_out<<<196, 256, 0, stream>>>(logits, mz, pgen, out);
}